// EdgesToNodesCollector_65249143161005
// MI455X (gfx1250) — compile-verified
//
#include <hip/hip_runtime.h>
#include <stdint.h>

// ---------------------------------------------------------------------------
// EdgesToNodesCollector for MI455X (gfx1250).
//
// Pure data movement (0 FLOPs): per node, copy the 4 incident edge rows
// (F=64 f32 = 256B each) into out[node] in ascending edge order, zero-padded.
// Traffic ~0.4 GB -> ~17us floor at 23.3 TB/s. WMMA is useless here; the
// CDNA5 feature we exploit is the async global<->LDS DMA path
// (GLOBAL_LOAD_ASYNC_TO_LDS_B128 / GLOBAL_STORE_ASYNC_FROM_LDS_B128, ASYNCcnt)
// with double-buffered LDS so stores of node k overlap loads of node k+1.
// ---------------------------------------------------------------------------

#define FEAT        64
#define ROW_BYTES   256          // FEAT * sizeof(float)
#define OUT_ROW     1024         // 4 * ROW_BYTES
#define MAXD        4
#define INVALID_ID  0xFFFFFFFFu

#if defined(__has_builtin)
#  if __has_builtin(__builtin_amdgcn_global_load_async_to_lds_b128) && \
      __has_builtin(__builtin_amdgcn_global_store_async_from_lds_b128)
#    define USE_ASYNC_BUILTINS 1
#  endif
#  if __has_builtin(__builtin_amdgcn_s_wait_asynccnt)
#    define HAVE_WAIT_ASYNC_BUILTIN 1
#  endif
#endif

// 128-bit payload type the async builtins are declared with.
typedef int v4i __attribute__((vector_size(4 * sizeof(int))));
typedef __attribute__((address_space(1))) v4i* gv4_t;   // global v4i*
typedef __attribute__((address_space(3))) v4i* lv4_t;   // LDS    v4i*

__device__ __forceinline__ void async_load_b128(uint64_t gaddr, uint32_t lds_byte) {
#if USE_ASYNC_BUILTINS
    __builtin_amdgcn_global_load_async_to_lds_b128(
        (gv4_t)(uintptr_t)gaddr,
        (lv4_t)(uintptr_t)lds_byte,
        0, 0);
#else
    asm volatile("global_load_async_to_lds_b128 %0, %1, off"
                 :: "v"(lds_byte), "v"(gaddr) : "memory");
#endif
}

__device__ __forceinline__ void async_store_b128(uint64_t gaddr, uint32_t lds_byte) {
#if USE_ASYNC_BUILTINS
    __builtin_amdgcn_global_store_async_from_lds_b128(
        (gv4_t)(uintptr_t)gaddr,
        (lv4_t)(uintptr_t)lds_byte,
        0, 0);
#else
    asm volatile("global_store_async_from_lds_b128 %0, %1, off"
                 :: "v"(gaddr), "v"(lds_byte) : "memory");
#endif
}

__device__ __forceinline__ void wait_async0() {
#if HAVE_WAIT_ASYNC_BUILTIN
    __builtin_amdgcn_s_wait_asynccnt(0);
#else
    asm volatile("s_wait_asynccnt 0x0" ::: "memory");
#endif
}

// ---------------------------------------------------------------------------
// Kernel 1: re-initialize workspace every call (ws is poisoned / stale).
// launch with >= 4*N threads.
// ---------------------------------------------------------------------------
__global__ __launch_bounds__(256)
void e2n_init_kernel(uint32_t* __restrict__ cnt, uint32_t* __restrict__ slots,
                     float* __restrict__ zrow, int N) {
    int i = blockIdx.x * blockDim.x + threadIdx.x;
    if (i < 4 * N) slots[i] = INVALID_ID;
    if (i < N)     cnt[i]   = 0u;
    if (i < FEAT)  zrow[i]  = 0.0f;
}

// ---------------------------------------------------------------------------
// Kernel 2: scatter incidences. One thread per (edge, endpoint).
// Self-loop: receiver side skipped (counted once), matching the reference.
// ---------------------------------------------------------------------------
__global__ __launch_bounds__(256)
void e2n_scatter_kernel(const int* __restrict__ senders,
                        const int* __restrict__ receivers,
                        uint32_t* __restrict__ cnt,
                        uint32_t* __restrict__ slots,
                        int E, int N) {
    int t = blockIdx.x * blockDim.x + threadIdx.x;
    if (t >= 2 * E) return;
    int e, node;
    if (t < E) {                       // sender endpoint
        e = t; node = senders[t];
    } else {                           // receiver endpoint
        e = t - E; node = receivers[e];
        if (senders[e] == node) return;    // self-loop counted once
    }
    if ((unsigned)node >= (unsigned)N) return;
    uint32_t slot = atomicAdd(&cnt[node], 1u);
    if (slot < MAXD) slots[node * MAXD + slot] = (uint32_t)e;
}

// ---------------------------------------------------------------------------
// Kernel 3: per-node 4-element sorting network -> ascending edge-index order.
// INVALID_ID (0xFFFFFFFF) padding sorts to the back automatically.
// ---------------------------------------------------------------------------
__device__ __forceinline__ void cswap(uint32_t& a, uint32_t& b) {
    uint32_t lo = (a < b) ? a : b;
    uint32_t hi = (a < b) ? b : a;
    a = lo; b = hi;
}

__global__ __launch_bounds__(256)
void e2n_sort_kernel(uint32_t* __restrict__ slots, int N) {
    int n = blockIdx.x * blockDim.x + threadIdx.x;
    if (n >= N) return;
    uint4 v = reinterpret_cast<uint4*>(slots)[n];
    cswap(v.x, v.y); cswap(v.z, v.w);
    cswap(v.x, v.z); cswap(v.y, v.w);
    cswap(v.y, v.z);
    reinterpret_cast<uint4*>(slots)[n] = v;
}

// ---------------------------------------------------------------------------
// Kernel 4: async-DMA gather/copy. One wave per node (grid-strided).
// Per node: 64 float4 chunks (1KB). Pass p in {0,1}: lane l handles chunk
// f = 32p + l; slot s = f>>4, byte off = (f&15)*16. Invalid slot -> zero row.
// Double-buffered 1KB LDS per wave: loads(k+1) overlap stores(k).
// ---------------------------------------------------------------------------
__global__ __launch_bounds__(256)
void e2n_gather_copy_kernel(const float* __restrict__ edges,
                            const uint32_t* __restrict__ slots,
                            const float* __restrict__ zrow,
                            float* __restrict__ out, int N) {
    __shared__ __align__(16) char lds[8 * 2 * OUT_ROW];   // 8 waves * 2 * 1KB

    const int lane      = threadIdx.x & 31;
    const int waveInBlk = threadIdx.x >> 5;
    const int wavesBlk  = blockDim.x >> 5;
    const int waveId    = blockIdx.x * wavesBlk + waveInBlk;
    const int nWaves    = gridDim.x * wavesBlk;

    const uint32_t myLdsBase =
        (uint32_t)(uintptr_t)(&lds[waveInBlk * 2 * OUT_ROW]);   // LDS byte addr
    const uint64_t edgesA = (uint64_t)(uintptr_t)edges;
    const uint64_t zrowA  = (uint64_t)(uintptr_t)zrow;
    const uint64_t outA   = (uint64_t)(uintptr_t)out;

    int k = 0;
    for (int node = waveId; node < N; node += nWaves, ++k) {
        const uint32_t buf = myLdsBase + (uint32_t)((k & 1) * OUT_ROW);

        // Issue 1KB of async loads (4 edge rows) into this wave's buffer.
        #pragma unroll
        for (int p = 0; p < 2; ++p) {
            const int f   = p * 32 + lane;           // float4 chunk 0..63
            const int s   = f >> 4;                  // slot 0..3
            const int off = (f & 15) << 4;           // byte offset in row
            const uint32_t eid = slots[node * MAXD + s];
            const uint64_t src = (eid != INVALID_ID)
                ? (edgesA + (uint64_t)eid * ROW_BYTES + (uint64_t)off)
                : (zrowA + (uint64_t)off);
            async_load_b128(src, buf + (uint32_t)(f << 4));
        }

        // Drain: this node's loads are in LDS (also drains prior stores).
        wait_async0();

        // Stream the 1KB row out; overlaps with next iteration's loads.
        #pragma unroll
        for (int p = 0; p < 2; ++p) {
            const int f = p * 32 + lane;
            const uint64_t dst = outA + (uint64_t)node * OUT_ROW
                                      + (uint64_t)(f << 4);
            async_store_b128(dst, buf + (uint32_t)(f << 4));
        }
    }
    wait_async0();   // all stores committed before wave retires
}

// ---------------------------------------------------------------------------
// Host-side launcher.
// d_in: [0]=nodes (N*64 f32, unused), [1]=edges (E*64 f32),
//       [2]=senders (E i32), [3]=receivers (E i32)
// d_out: N * 256 f32.
// ws layout: cnt[N] u32 | slots[4N] u32 | zrow[64] f32   (~5 MB)
// ---------------------------------------------------------------------------
extern "C" void kernel_launch(void* const* d_in, const int* in_sizes, int n_in,
                              void* d_out, int out_size, void* d_ws, size_t ws_size,
                              hipStream_t stream) {
    (void)n_in; (void)out_size; (void)ws_size;

    const int N = in_sizes[0] / FEAT;       // 262144
    const int E = in_sizes[2];              // 524288

    const float* edges     = (const float*)d_in[1];
    const int*   senders   = (const int*)d_in[2];
    const int*   receivers = (const int*)d_in[3];
    float*       out       = (float*)d_out;

    uint32_t* cnt   = (uint32_t*)d_ws;
    uint32_t* slots = cnt + N;
    float*    zrow  = (float*)(slots + 4 * (size_t)N);

    const int T = 256;
    e2n_init_kernel<<<(4 * N + T - 1) / T, T, 0, stream>>>(cnt, slots, zrow, N);
    e2n_scatter_kernel<<<(2 * E + T - 1) / T, T, 0, stream>>>(
        senders, receivers, cnt, slots, E, N);
    e2n_sort_kernel<<<(N + T - 1) / T, T, 0, stream>>>(slots, N);

    // 4096 blocks * 8 waves = 32768 waves, 8 nodes each: enough concurrency
    // to keep 23.3 TB/s of HBM busy while each wave is a 2-deep DMA pipeline.
    int blocks = (N / (8 * (T / 32)));      // N / 64 = 4096
    if (blocks < 1) blocks = 1;
    e2n_gather_copy_kernel<<<blocks, T, 0, stream>>>(edges, slots, zrow, out, N);
}